// Model_11879879543238
// MI455X (gfx1250) — compile-verified
//
#include <hip/hip_runtime.h>

#define KSZ   22
#define ORG   11               // pad-before (kornia origin = K/2)
#define TW    64               // output tile width
#define TH    64               // output tile height
#define IN_W  (TW + KSZ - 1)   // 85
#define IN_H  (TH + KSZ - 1)   // 85
#define C_W   (TW + 16)        // 80: m4 taps reach c+16
#define NT    256
#define IMG_W 1024
#define IMG_H 1024

typedef unsigned int       u32;
typedef unsigned long long u64;
typedef __attribute__((ext_vector_type(4))) unsigned int u32x4;
typedef __attribute__((ext_vector_type(4))) int          i32x4;
typedef __attribute__((ext_vector_type(8))) int          i32x8;

#if defined(__has_builtin)
#if __has_builtin(__builtin_amdgcn_tensor_load_to_lds) && \
    __has_builtin(__builtin_amdgcn_tensor_store_from_lds) && \
    __has_builtin(__builtin_amdgcn_s_wait_tensorcnt)
#define HAVE_TDM 1
#endif
#endif
#ifndef HAVE_TDM
#define HAVE_TDM 0
#endif

#if HAVE_TDM
#if __has_include(<hip/amd_detail/amd_gfx1250_TDM.h>)
#define TDM_ARITY6 1
#else
#define TDM_ARITY6 0
#endif

// Build a 2-D f32 tile D# and issue tensor_load_to_lds / tensor_store_from_lds.
// tile: tile_w x tile_h elements, global row stride stride_elems, contiguous in
// LDS at lds_byte_off.  tdim0/tdim1 = tensor extents from global_addr for the
// hardware OOB check (loads beyond -> zero-filled, stores beyond -> dropped).
template <bool IS_STORE>
__device__ __forceinline__ void tdm_issue(u64 ga, u32 lds_byte_off,
                                          int tile_w, int tile_h,
                                          u32 tdim0, u32 tdim1, u32 stride_elems)
{
    u32x4 g0;
    g0.x = 1u;                                              // count=1, user mode
    g0.y = lds_byte_off;                                    // lds_addr [63:32]
    g0.z = (u32)ga;                                         // global_addr lo32
    g0.w = (u32)((ga >> 32) & 0x01FFFFFFull) | (2u << 30);  // ga[56:32] | type=2

    i32x8 g1;
    g1[0] = (int)(2u << 16);                                // wg_mask=0, data_size=4B
    g1[1] = (int)((tdim0 & 0xFFFFu) << 16);                 // tensor_dim0[15:0]
    g1[2] = (int)((tdim0 >> 16) | ((tdim1 & 0xFFFFu) << 16));
    g1[3] = (int)((tdim1 >> 16) | ((u32)tile_w << 16));     // tile_dim0
    g1[4] = (int)tile_h;                                    // tile_dim1, tile_dim2=0
    g1[5] = (int)stride_elems;                              // tensor_dim0_stride lo
    g1[6] = 0;
    g1[7] = 0;

    i32x4 gz = {0, 0, 0, 0};                                // groups 2/3: 2-D only
#if TDM_ARITY6
    i32x8 gz8 = {0, 0, 0, 0, 0, 0, 0, 0};
    if (IS_STORE) __builtin_amdgcn_tensor_store_from_lds(g0, g1, gz, gz, gz8, 0);
    else          __builtin_amdgcn_tensor_load_to_lds  (g0, g1, gz, gz, gz8, 0);
#else
    if (IS_STORE) __builtin_amdgcn_tensor_store_from_lds(g0, g1, gz, gz, 0);
    else          __builtin_amdgcn_tensor_load_to_lds  (g0, g1, gz, gz, 0);
#endif
}
#endif // HAVE_TDM

// One separable 22-tap morphology pass (min if !IS_MAX, max if IS_MAX).
// Per direction:  m4[c] = op(x[c..c+3])           (3 ops, direct from input)
//                 win22[c] = op(m4[c], m4[c+4], m4[c+8], m4[c+12], m4[c+16],
//                               op(x[c+20], x[c+21]))
// which covers exactly [c, c+21] (min/max are idempotent under overlap).
template <bool IS_MAX>
__global__ __launch_bounds__(NT)
void morph_pass(const float* __restrict__ src, float* __restrict__ dst)
{
    __shared__ float A [IN_H * IN_W];   // input tile; later m4v (flat stride TW)
    __shared__ float Cx[IN_H * C_W];    // m4h; later output staging (TW x TH)
    __shared__ float Bx[IN_H * TW];     // horizontal win22 result

    const int tid   = threadIdx.x;
    const int tx    = blockIdx.x;
    const int ty    = blockIdx.y;
    const int plane = blockIdx.z;

    const long long pbase = (long long)plane * (IMG_W * IMG_H);
    const float* __restrict__ p = src + pbase;
    float*       __restrict__ q = dst + pbase;

    const int gx0 = tx * TW - ORG;
    const int gy0 = ty * TH - ORG;
    const float PADV = IS_MAX ? -__builtin_inff() : __builtin_inff();

    const bool interior = (gx0 >= 0) && (gy0 >= 0) &&
                          (gx0 + IN_W <= IMG_W) && (gy0 + IN_H <= IMG_H);

    // ---------------- tile load: TDM DMA (interior) or guarded ----------------
#if HAVE_TDM
    if (interior) {
        const int w = tid >> 5;          // wave index
        if (w < 2) {                     // two DMAs in flight (rows split 43/42)
            const int r0 = w * 43;
            const int nr = (w == 0) ? 43 : (IN_H - 43);
            tdm_issue<false>((u64)(const void*)(p + (long long)(gy0 + r0) * IMG_W + gx0),
                             (u32)(u64)(void*)A + (u32)(r0 * IN_W * 4),
                             IN_W, nr, 0x40000000u, 0x40000000u, IMG_W);
            __builtin_amdgcn_s_wait_tensorcnt(0);
        }
    } else
#else
    (void)interior;
#endif
    {
        for (int i = tid; i < IN_H * IN_W; i += NT) {
            int r = i / IN_W, c = i - r * IN_W;
            int gy = gy0 + r, gx = gx0 + c;
            float v = PADV;
            if ((unsigned)gy < (unsigned)IMG_H && (unsigned)gx < (unsigned)IMG_W)
                v = p[gy * IMG_W + gx];
            A[i] = v;
        }
    }
    __syncthreads();

    auto OP = [](float a, float b) { return IS_MAX ? fmaxf(a, b) : fminf(a, b); };

    // ---- horizontal m4: Cx[r][c] = op(A[r][c..c+3]), c < 80, all 85 rows ----
    for (int i = tid; i < IN_H * C_W; i += NT) {
        int r = i / C_W, c = i - r * C_W;
        const float* Ar = &A[r * IN_W + c];
        Cx[i] = OP(OP(Ar[0], Ar[1]), OP(Ar[2], Ar[3]));
    }
    __syncthreads();

    // ---- horizontal win22 -> Bx (width TW), m2 tail recomputed from A ----
    for (int i = tid; i < IN_H * TW; i += NT) {
        int r = i / TW, c = i - r * TW;
        const float* Cr = &Cx[r * C_W + c];
        const float* Ar = &A [r * IN_W + c];
        Bx[i] = OP(OP(OP(Cr[0], Cr[4]), OP(Cr[8], Cr[12])),
                   OP(Cr[16], OP(Ar[20], Ar[21])));
    }
    __syncthreads();

    // ---- vertical m4 -> A (flat, stride TW), rows 0..79 ----
    for (int i = tid; i < (TH + 16) * TW; i += NT) {
        const float* Br = &Bx[i];
        A[i] = OP(OP(Br[0], Br[TW]), OP(Br[2 * TW], Br[3 * TW]));
    }
    __syncthreads();

    // ---- vertical win22, m2 tail from Bx; stage result / store ----
    for (int i = tid; i < TH * TW; i += NT) {
        const float* Ar = &A [i];
        const float* Br = &Bx[i];
        float v = OP(OP(OP(Ar[0], Ar[4 * TW]), OP(Ar[8 * TW], Ar[12 * TW])),
                     OP(Ar[16 * TW], OP(Br[20 * TW], Br[21 * TW])));
#if HAVE_TDM
        Cx[i] = v;                               // stage tile for DMA store
#else
        int r = i / TW, c = i - r * TW;
        q[(long long)(ty * TH + r) * IMG_W + (tx * TW + c)] = v;
#endif
    }

#if HAVE_TDM
    __syncthreads();
    if (tid < 32) {                              // wave 0 emits the tile DMA
        const int ox = tx * TW, oy = ty * TH;
        tdm_issue<true>((u64)(void*)(q + (long long)oy * IMG_W + ox),
                        (u32)(u64)(void*)Cx,
                        TW, TH, (u32)(IMG_W - ox), (u32)(IMG_H - oy), IMG_W);
        __builtin_amdgcn_s_wait_tensorcnt(0);
    }
#endif
}

extern "C" void kernel_launch(void* const* d_in, const int* in_sizes, int n_in,
                              void* d_out, int out_size, void* d_ws, size_t ws_size,
                              hipStream_t stream)
{
    const float* img = (const float*)d_in[0];
    float*       out = (float*)d_out;
    float*       tmp = (float*)d_ws;   // erosion intermediate (96 MB)

    const int planes = in_sizes[0] / (IMG_W * IMG_H);  // 8*3 = 24

    dim3 grid(IMG_W / TW, IMG_H / TH, planes);         // 16 x 16 x 24
    dim3 block(NT);

    // opening = dilate(erode(x)); each pass is one fused separable kernel
    hipLaunchKernelGGL((morph_pass<false>), grid, block, 0, stream, img, tmp); // erosion  (min, pad +inf)
    hipLaunchKernelGGL((morph_pass<true>),  grid, block, 0, stream, tmp, out); // dilation (max, pad -inf)
}